// Router_76871324663966
// MI455X (gfx1250) — compile-verified
//
#include <hip/hip_runtime.h>
#include <hip/hip_bf16.h>
#include <math.h>

typedef float v2f __attribute__((ext_vector_type(2)));
typedef float v4f __attribute__((ext_vector_type(4)));
typedef float v8f __attribute__((ext_vector_type(8)));

#define G_DIM   4
#define S_DIM   2048
#define D_DIM   2048
#define E_DIM   64
#define TOKENS  (G_DIM * S_DIM)          // 8192
#define CAP     64
#define CSLOTS  (CAP - 1)                // 63 (capacity slot 0 dropped)
#define SLOTS   (E_DIM * CSLOTS)         // 4032 per token
#define NCOMB   ((size_t)TOKENS * SLOTS) // 33,030,144 floats per output tensor
#define KCHUNK  128

// ---------------------------------------------------------------------------
// K1: logits = x @ W  (bias added later).  M=8192, K=2048, N=64, fp32 WMMA.
// One wave per 16-row strip, 4 accumulators cover all 64 output columns.
// W chunks staged through LDS (32KB); x read exactly once from HBM.
// ---------------------------------------------------------------------------
__global__ __launch_bounds__(128) void router_gemm_wmma(
    const float* __restrict__ X, const float* __restrict__ Wm,
    float* __restrict__ logits)
{
  __shared__ float wlds[KCHUNK * E_DIM];           // 32 KB
  const int tid  = threadIdx.x;
  const int wave = tid >> 5;                       // 0..3
  const int lane = tid & 31;
  const int half = lane >> 4;                      // 0/1 (K-pair select)
  const int mrow = lane & 15;                      // M for A, N for B/C
  const int strip = blockIdx.x * 4 + wave;         // 0..511
  const int row   = strip * 16 + mrow;
  const float* xrow = X + (size_t)row * D_DIM + 2 * half;

  v8f acc0 = {}, acc1 = {}, acc2 = {}, acc3 = {};

  for (int kc = 0; kc < D_DIM; kc += KCHUNK) {
    // cooperative 128x64 W chunk load -> LDS (b128, fully coalesced)
    {
      const v4f* wsrc = (const v4f*)(Wm + (size_t)kc * E_DIM);
      v4f* wdst = (v4f*)wlds;
      #pragma unroll
      for (int i = 0; i < (KCHUNK * E_DIM) / 4 / 128; ++i)
        wdst[tid + i * 128] = wsrc[tid + i * 128];
    }
    __syncthreads();
    // hint next x chunk into cache (global_prefetch_b8, speculative)
    __builtin_prefetch(xrow + kc + KCHUNK, 0, 0);

    #pragma unroll 4
    for (int kk = 0; kk < KCHUNK; kk += 4) {
      // A 16x4 f32: lanes 0-15 K={kk,kk+1}, lanes 16-31 K={kk+2,kk+3}
      v2f a;
      a.x = xrow[kc + kk + 0];
      a.y = xrow[kc + kk + 1];
      const float* wb = wlds + (kk + 2 * half) * E_DIM + mrow;
      v2f b0, b1, b2, b3;
      b0.x = wb[0];  b0.y = wb[0  + E_DIM];
      b1.x = wb[16]; b1.y = wb[16 + E_DIM];
      b2.x = wb[32]; b2.y = wb[32 + E_DIM];
      b3.x = wb[48]; b3.y = wb[48 + E_DIM];
      acc0 = __builtin_amdgcn_wmma_f32_16x16x4_f32(false, a, false, b0, (short)0, acc0, false, false);
      acc1 = __builtin_amdgcn_wmma_f32_16x16x4_f32(false, a, false, b1, (short)0, acc1, false, false);
      acc2 = __builtin_amdgcn_wmma_f32_16x16x4_f32(false, a, false, b2, (short)0, acc2, false, false);
      acc3 = __builtin_amdgcn_wmma_f32_16x16x4_f32(false, a, false, b3, (short)0, acc3, false, false);
    }
    __syncthreads();
  }

  // C/D layout: acc[r] holds M = r + 8*half, N = n*16 + mrow
  #pragma unroll
  for (int r = 0; r < 8; ++r) {
    const int orow = strip * 16 + r + half * 8;
    float* o = logits + (size_t)orow * E_DIM + mrow;
    o[0]  = acc0[r];
    o[16] = acc1[r];
    o[32] = acc2[r];
    o[48] = acc3[r];
  }
}

// ---------------------------------------------------------------------------
// K2: per-token softmax over E=64 + top-2.  top-k(softmax) == top-k(logits);
// gates from exp-sum.  Strict '>' reproduces jax top_k tie-breaking.
// ---------------------------------------------------------------------------
__global__ __launch_bounds__(256) void router_softmax_top2(
    const float* __restrict__ logits, const float* __restrict__ bias,
    float* __restrict__ gates, int* __restrict__ idx)
{
  const int t = blockIdx.x * 256 + threadIdx.x;
  if (t >= TOKENS) return;
  const float* l = logits + (size_t)t * E_DIM;

  float m0 = -__builtin_inff(), m1 = -__builtin_inff();
  int i0 = 0, i1 = 0;
  #pragma unroll 4
  for (int e = 0; e < E_DIM; ++e) {
    const float v = l[e] + bias[e];
    if (v > m0)      { m1 = m0; i1 = i0; m0 = v; i0 = e; }
    else if (v > m1) { m1 = v;  i1 = e; }
  }
  float s = 0.f;
  #pragma unroll 4
  for (int e = 0; e < E_DIM; ++e)
    s += __expf(l[e] + bias[e] - m0);

  const float inv = 1.0f / s;
  gates[2 * t + 0] = inv;                   // exp(m0-m0)/s
  gates[2 * t + 1] = __expf(m1 - m0) * inv;
  idx[2 * t + 0] = i0;
  idx[2 * t + 1] = i1;
}

// ---------------------------------------------------------------------------
// K3: sequential cumsum over S, independently per (g, k, e).
// One thread per (k,e) pair; 4 blocks (one per group).
// ---------------------------------------------------------------------------
__global__ __launch_bounds__(128) void router_positions(
    const int* __restrict__ idx, int* __restrict__ pos)
{
  const int g = blockIdx.x;
  const int k = threadIdx.x >> 6;   // 0..1
  const int e = threadIdx.x & 63;   // 0..63
  int cnt = 0;
  const int base = g * S_DIM;
  for (int s = 0; s < S_DIM; ++s) {
    const int off = (base + s) * 2 + k;
    if (idx[off] == e) { ++cnt; pos[off] = cnt; }
  }
}

// ---------------------------------------------------------------------------
// K4: dense output writer (264 MB stream -> dominant cost).
// One block per token; branch-free value per slot; NT b128 stores.
// out = [combine (TOKENS*4032) | mask (TOKENS*4032) | loss (1)]
// ---------------------------------------------------------------------------
__global__ __launch_bounds__(256) void router_writeout(
    const float* __restrict__ gates, const int* __restrict__ idx,
    const int* __restrict__ pos, float* __restrict__ out)
{
  const int t = blockIdx.x;
  const float g0 = gates[2 * t + 0];
  const float g1 = gates[2 * t + 1];
  const int   i0 = idx[2 * t + 0],  i1 = idx[2 * t + 1];
  const int   p0 = pos[2 * t + 0],  p1 = pos[2 * t + 1];

  v4f* comb = (v4f*)(out + (size_t)t * SLOTS);
  v4f* mask = (v4f*)(out + NCOMB + (size_t)t * SLOTS);

  for (int j = threadIdx.x; j < SLOTS / 4; j += 256) {
    v4f c, m;
    const int s4 = 4 * j;
    #pragma unroll
    for (int q = 0; q < 4; ++q) {
      const int slot = s4 + q;
      const int e  = slot / CSLOTS;          // expert
      const int cc = slot - e * CSLOTS;      // capacity slot (0..62) == pos-1
      float val = 0.f;
      if (i0 == e && p0 == cc + 1) val  = g0;
      if (i1 == e && p1 == cc + 1) val += g1;
      c[q] = val;
      m[q] = (val != 0.f) ? 1.f : 0.f;
    }
    __builtin_nontemporal_store(c, comb + j);
    __builtin_nontemporal_store(m, mask + j);
  }
  if (t == 0 && threadIdx.x == 0) out[2 * NCOMB] = 0.0f;   // loss
}

// ---------------------------------------------------------------------------
extern "C" void kernel_launch(void* const* d_in, const int* in_sizes, int n_in,
                              void* d_out, int out_size, void* d_ws, size_t ws_size,
                              hipStream_t stream) {
  const float* X    = (const float*)d_in[0];  // (4,2048,2048)
  const float* Wm   = (const float*)d_in[1];  // (2048,64)
  const float* bias = (const float*)d_in[2];  // (64,)
  float* out = (float*)d_out;

  // workspace: logits | gates | idx | pos  (~2.2 MB)
  float* logits = (float*)d_ws;
  float* gates  = logits + (size_t)TOKENS * E_DIM;
  int*   idx    = (int*)(gates + 2 * TOKENS);
  int*   pos    = idx + 2 * TOKENS;

  router_gemm_wmma   <<<TOKENS / 16 / 4, 128, 0, stream>>>(X, Wm, logits);
  router_softmax_top2<<<TOKENS / 256,    256, 0, stream>>>(logits, bias, gates, idx);
  router_positions   <<<G_DIM,           128, 0, stream>>>(idx, pos);
  router_writeout    <<<TOKENS,          256, 0, stream>>>(gates, idx, pos, out);
}